// AutoEncoder_14542759264279
// MI455X (gfx1250) — compile-verified
//
#include <hip/hip_runtime.h>

#define XS    128
#define VOX   (XS * XS * XS)        // 2,097,152 voxels per batch
#define BATCH 8
#define MPTS  500000                // divisible by 4

typedef float v2f __attribute__((ext_vector_type(2)));
typedef float v4f __attribute__((ext_vector_type(4)));
typedef float v8f __attribute__((ext_vector_type(8)));
typedef int   v4i __attribute__((ext_vector_type(4)));

// ---------------------------------------------------------------------------
// Kernel 1: zero the 64 MB grid workspace (float4 stores) and the 16-float out.
// Regular-temporal stores: we WANT the grid resident in L2 for the scatter.
// ---------------------------------------------------------------------------
__global__ void zero_kernel(float* __restrict__ grid, float* __restrict__ out) {
    const size_t i = (size_t)blockIdx.x * blockDim.x + threadIdx.x; // 16384*256 float4 = 64 MB
    float4* g4 = (float4*)grid;
    g4[i] = make_float4(0.f, 0.f, 0.f, 0.f);
    if (blockIdx.x == 0 && threadIdx.x < 16) out[threadIdx.x] = 0.0f;
}

// ---------------------------------------------------------------------------
// Kernel 2: scatter-add. 4 points per thread:
//   3 x b128 NT index loads (48 B) + 1 x b128 NT value load (16 B),
//   4 x global_atomic_add_f32 into the L2-resident grid.
// NT hints keep the 64 MB one-pass stream from evicting the grid in L2.
// MPTS % 4 == 0, so all 4 points of a thread share one batch.
// ---------------------------------------------------------------------------
__global__ void scatter_kernel(const int* __restrict__ idx,
                               const float* __restrict__ val,
                               float* __restrict__ grid) {
    const unsigned t = blockIdx.x * blockDim.x + threadIdx.x;   // 1,000,000 threads
    if (t >= (BATCH * MPTS) / 4) return;

    const v4i* i4 = (const v4i*)(idx + (size_t)t * 12);
    const v4i a = __builtin_nontemporal_load(i4 + 0);  // x0 y0 z0 x1
    const v4i b = __builtin_nontemporal_load(i4 + 1);  // y1 z1 x2 y2
    const v4i c = __builtin_nontemporal_load(i4 + 2);  // z2 x3 y3 z3
    const v4f v = __builtin_nontemporal_load((const v4f*)(val + (size_t)t * 4));

    const unsigned bb = t / (MPTS / 4);
    float* g = grid + (size_t)bb * VOX;

    atomicAdd(g + ((((size_t)a[0] * XS + a[1]) * XS) + a[2]), v[0]);
    atomicAdd(g + ((((size_t)a[3] * XS + b[0]) * XS) + b[1]), v[1]);
    atomicAdd(g + ((((size_t)b[2] * XS + b[3]) * XS) + c[0]), v[2]);
    atomicAdd(g + ((((size_t)c[1] * XS + c[2]) * XS) + c[3]), v[3]);
}

// ---------------------------------------------------------------------------
// Wave32 sum via V_WMMA_F32_16X16X4_F32.
// A (16x4 f32, 2 VGPRs/lane): lanes 0-15 hold {K0,K1}, lanes 16-31 hold {K2,K3}.
// a = {v, 0} -> A[m][0]=v_m, A[m][2]=v_{m+16}, rest 0.
// B = all-ones (layout-invariant) -> D[m][n] = v_m + v_{m+16}.
// Sum the 8 D VGPRs per lane, then lane0 + lane16 = full 32-lane sum.
// ---------------------------------------------------------------------------
__device__ __forceinline__ float wave_sum_wmma(float v) {
    v2f a;  a[0] = v;    a[1] = 0.0f;
    v2f bm; bm[0] = 1.0f; bm[1] = 1.0f;
    v8f c = {};
    v8f d = __builtin_amdgcn_wmma_f32_16x16x4_f32(
        /*neg_a=*/false, a, /*neg_b=*/false, bm,
        /*c_mod=*/(short)0, c, /*reuse_a=*/false, /*reuse_b=*/false);
    float s = ((d[0] + d[1]) + (d[2] + d[3])) + ((d[4] + d[5]) + (d[6] + d[7]));
    return __shfl(s, 0, 32) + __shfl(s, 16, 32);
}

// ---------------------------------------------------------------------------
// Kernel 3: TV / MSE reduction. One thread per (b,x,y) z-row (131072 rows).
// Linear voxel index = row*128 + z: d3 is register-local within the row,
// d2 neighbor row is +128 floats, d1 neighbor row is +16384 floats (L2 hits).
// Each block (256 threads) lies entirely within one batch (16384 % 256 == 0).
// ---------------------------------------------------------------------------
__global__ void tv_kernel(const float* __restrict__ grid, float* __restrict__ out) {
    const unsigned tid = threadIdx.x;
    const unsigned r   = blockIdx.x * 256u + tid;   // row id = b*16384 + x*128 + y
    const unsigned y   = r & 127u;
    const unsigned x   = (r >> 7) & 127u;
    const bool hasY = (y < 127u);
    const bool hasX = (x < 127u);

    const float4* c4 = (const float4*)(grid + (size_t)r * XS);
    const float4* y4 = c4 + (XS / 4);               // +128 floats
    const float4* x4 = c4 + (XS * XS / 4);          // +16384 floats

    float tv = 0.f, mse = 0.f;
    float prev = 0.f;
    #pragma unroll 4
    for (int j = 0; j < 32; ++j) {
        const float4 c = c4[j];
        // d3: diffs along contiguous z axis
        if (j > 0) { const float d = c.x - prev; tv += fabsf(d); mse += d * d; }
        { const float d = c.y - c.x; tv += fabsf(d); mse += d * d; }
        { const float d = c.z - c.y; tv += fabsf(d); mse += d * d; }
        { const float d = c.w - c.z; tv += fabsf(d); mse += d * d; }
        prev = c.w;
        // d2: y-axis neighbor row
        if (hasY) {
            const float4 n = y4[j];
            const float d0 = n.x - c.x, d1 = n.y - c.y, d2 = n.z - c.z, d3 = n.w - c.w;
            tv  += fabsf(d0) + fabsf(d1) + fabsf(d2) + fabsf(d3);
            mse += d0 * d0 + d1 * d1 + d2 * d2 + d3 * d3;
        }
        // d1: x-axis neighbor row
        if (hasX) {
            const float4 n = x4[j];
            const float d0 = n.x - c.x, d1 = n.y - c.y, d2 = n.z - c.z, d3 = n.w - c.w;
            tv  += fabsf(d0) + fabsf(d1) + fabsf(d2) + fabsf(d3);
            mse += d0 * d0 + d1 * d1 + d2 * d2 + d3 * d3;
        }
    }

    // No early returns above -> EXEC is all-ones here (WMMA requirement).
    const float wtv  = wave_sum_wmma(tv);
    const float wmse = wave_sum_wmma(mse);

    if ((tid & 31u) == 0u) {
        const unsigned b = (blockIdx.x * 256u) >> 14;                 // uniform per block
        const float inv_pix   = 1.0f / (float)VOX;                    // 1 / X^3
        const float inv_pairs = 1.0f / (float)(2 * XS * XS - 2 * XS); // 1 / 32512
        atomicAdd(&out[b],     wtv  * inv_pix);
        atomicAdd(&out[8 + b], wmse * inv_pairs);
    }
}

// ---------------------------------------------------------------------------
extern "C" void kernel_launch(void* const* d_in, const int* in_sizes, int n_in,
                              void* d_out, int out_size, void* d_ws, size_t ws_size,
                              hipStream_t stream) {
    const int*   indices = (const int*)d_in[0];   // (B, M, 3) int32
    const float* values  = (const float*)d_in[1]; // (B, M) float32
    // d_in[2] = xsize scalar (128), compile-time constant here.

    float* grid = (float*)d_ws;                   // needs 8 * 128^3 * 4 = 64 MB
    float* out  = (float*)d_out;                  // 16 floats: [tv(8) | mse(8)]

    // 1) zero grid + out
    zero_kernel<<<16384, 256, 0, stream>>>(grid, out);
    // 2) scatter-add 4M points, 4 per thread (NT streamed loads, L2 atomics)
    scatter_kernel<<<((BATCH * MPTS) / 4 + 255) / 256, 256, 0, stream>>>(indices, values, grid);
    // 3) diff reductions, WMMA wave reduce, scaled atomics into out
    tv_kernel<<<(BATCH * XS * XS) / 256, 256, 0, stream>>>(grid, out);
}